// MACEEncoder_79714593014002
// MI455X (gfx1250) — compile-verified
//
#include <hip/hip_runtime.h>
#include <math.h>

#define N_ATOMS  80000
#define N_BLK    20000
#define NE       320000
#define C        128
#define N_RBF    16
#define HR       64
#define NL       2
#define N_GR     512
#define R_MAX_F  5.0f
#define PI_F     3.14159265358979323846f
#define MBLK     5            // M-tiles (16 rows each) per wave in node GEMMs
#define MSG_WAVES 4           // waves per block in fused message kernel

typedef float v2f __attribute__((ext_vector_type(2)));
typedef float v4f __attribute__((ext_vector_type(4)));
typedef float v8f __attribute__((ext_vector_type(8)));

// ---------------------------------------------------------------------------
// V_WMMA_F32_16X16X4_F32 wrapper: D(16x16 f32) = A(16x4) * B(4x16) + C
// ---------------------------------------------------------------------------
__device__ __forceinline__ v8f wmma4(v2f a, v2f b, v8f c) {
    return __builtin_amdgcn_wmma_f32_16x16x4_f32(false, a, false, b,
                                                 (short)0, c, false, false);
}

__device__ __forceinline__ v2f loadA(const float* A, int ld, int row0, int k0,
                                     int lm, int lg) {
    const float* p = A + (size_t)(row0 + lm) * ld + k0 + 2 * lg;
    v2f a; a.x = p[0]; a.y = p[1]; return a;
}
__device__ __forceinline__ v2f loadB(const float* B, int ld, int k0, int n0,
                                     int lm, int lg) {
    const float* p = B + (size_t)(k0 + 2 * lg) * ld + n0 + lm;
    v2f b; b.x = p[0]; b.y = p[ld]; return b;
}

// ---------------------------------------------------------------------------
// 1) Segment mean: atomic accumulate then normalize
// ---------------------------------------------------------------------------
__global__ __launch_bounds__(128) void seg_acc_kernel(
    const float* __restrict__ H, const float* __restrict__ Z,
    const int* __restrict__ block_id, float* __restrict__ Hb,
    float* __restrict__ Zb, float* __restrict__ cnt) {
    int a = blockIdx.x;
    int c = threadIdx.x;
    int b = block_id[a];
    atomicAdd(&Hb[(size_t)b * C + c], H[(size_t)a * C + c]);
    if (c < 3) atomicAdd(&Zb[(size_t)b * 3 + c], Z[(size_t)a * 3 + c]);
    if (c == 0) atomicAdd(&cnt[b], 1.0f);
}

__global__ __launch_bounds__(128) void seg_norm_kernel(
    float* __restrict__ Hb, float* __restrict__ Zb,
    const float* __restrict__ cnt) {
    int b = blockIdx.x;
    int c = threadIdx.x;
    float inv = 1.0f / fmaxf(cnt[b], 1.0f);
    Hb[(size_t)b * C + c] *= inv;
    if (c < 3) Zb[(size_t)b * 3 + c] *= inv;
}

// ---------------------------------------------------------------------------
// 2) Edge geometry: unit vector + (rbf * fcut) features
// ---------------------------------------------------------------------------
__global__ __launch_bounds__(256) void edge_kernel(
    const float* __restrict__ Zb, const int* __restrict__ src,
    const int* __restrict__ dst, float* __restrict__ u,
    float* __restrict__ rfeat) {
    int e = blockIdx.x * 256 + threadIdx.x;
    if (e >= NE) return;
    int s = src[e], d_ = dst[e];
    float rx = Zb[(size_t)d_ * 3 + 0] - Zb[(size_t)s * 3 + 0];
    float ry = Zb[(size_t)d_ * 3 + 1] - Zb[(size_t)s * 3 + 1];
    float rz = Zb[(size_t)d_ * 3 + 2] - Zb[(size_t)s * 3 + 2];
    float d = sqrtf(rx * rx + ry * ry + rz * rz + 1e-12f);
    float inv = 1.0f / d;
    u[(size_t)e * 3 + 0] = rx * inv;
    u[(size_t)e * 3 + 1] = ry * inv;
    u[(size_t)e * 3 + 2] = rz * inv;
    float x = d * (1.0f / R_MAX_F);
    float x2 = x * x, x4 = x2 * x2, x5 = x4 * x, x6 = x5 * x, x7 = x6 * x;
    float fcut = (x < 1.0f) ? (1.0f - 21.0f * x5 + 35.0f * x6 - 15.0f * x7) : 0.0f;
    float pref = sqrtf(2.0f / R_MAX_F) * inv * fcut;
    float base = PI_F * d * (1.0f / R_MAX_F);
    #pragma unroll
    for (int n = 1; n <= N_RBF; n++)
        rfeat[(size_t)e * N_RBF + n - 1] = pref * __sinf((float)n * base);
}

// ---------------------------------------------------------------------------
// 3) h_s = Hb @ W_embed   (WMMA f32, 80x16 output strip per wave)
// ---------------------------------------------------------------------------
__global__ __launch_bounds__(32) void gemm_embed_kernel(
    const float* __restrict__ A, const float* __restrict__ B,
    float* __restrict__ Cout) {
    int lane = threadIdx.x & 31, lm = lane & 15, lg = lane >> 4;
    int row0 = blockIdx.x * (16 * MBLK), n0 = blockIdx.y * 16;
    v8f acc[MBLK];
    #pragma unroll
    for (int m = 0; m < MBLK; m++)
        acc[m] = (v8f){0.f, 0.f, 0.f, 0.f, 0.f, 0.f, 0.f, 0.f};
    for (int kk = 0; kk < C / 4; kk++) {
        int k0 = kk * 4;
        v2f b = loadB(B, C, k0, n0, lm, lg);
        #pragma unroll
        for (int m = 0; m < MBLK; m++)
            acc[m] = wmma4(loadA(A, C, row0 + 16 * m, k0, lm, lg), b, acc[m]);
    }
    #pragma unroll
    for (int m = 0; m < MBLK; m++)
        #pragma unroll
        for (int j = 0; j < 8; j++)
            Cout[(size_t)(row0 + 16 * m + j + 8 * lg) * C + n0 + lm] = acc[m][j];
}

// ---------------------------------------------------------------------------
// 4) Fused radial MLP + message scatter.  4 waves/block, 16 edges/wave.
//    Weights staged once per block into LDS (b128 copies), stage-2 B
//    fragments are served from LDS.
// ---------------------------------------------------------------------------
__global__ __launch_bounds__(32 * MSG_WAVES) void msg_kernel(
    const float* __restrict__ rfeat, const float* __restrict__ u,
    const int* __restrict__ src, const int* __restrict__ dst,
    const float* __restrict__ Wr1l, const float* __restrict__ br1l,
    const float* __restrict__ Wr2l, const float* __restrict__ hs,
    const float* __restrict__ hv, float* __restrict__ agg_s,
    float* __restrict__ agg_v) {
    __shared__ float wr1s[N_RBF * HR];               //   4 KB
    __shared__ float br1s[HR];                       //  256 B
    __shared__ float wr2s[HR * 2 * C];               //  64 KB
    __shared__ float hid[MSG_WAVES][16 * HR];        //  16 KB
    __shared__ float wt[MSG_WAVES][16 * 2 * C];      //  64 KB
    int tid = threadIdx.x;
    int wave = tid >> 5;
    int lane = tid & 31;
    int lm = lane & 15, lg = lane >> 4;
    int e0 = (blockIdx.x * MSG_WAVES + wave) * 16;

    // ---- cooperative weight staging (vectorized b128) ----
    {
        const v4f* g2 = (const v4f*)Wr2l;
        v4f* s2 = (v4f*)wr2s;
        for (int i = tid; i < (HR * 2 * C) / 4; i += 32 * MSG_WAVES)
            s2[i] = g2[i];
        const v4f* g1 = (const v4f*)Wr1l;
        v4f* s1 = (v4f*)wr1s;
        for (int i = tid; i < (N_RBF * HR) / 4; i += 32 * MSG_WAVES)
            s1[i] = g1[i];
        if (tid < HR) br1s[tid] = br1l[tid];
    }
    // prefetch gather rows for stage 3
    if (lane < 16) {
        int sp = src[e0 + lane];
        __builtin_prefetch(hs + (size_t)sp * C, 0, 0);
        __builtin_prefetch(hv + (size_t)sp * C * 3, 0, 0);
    }
    __syncthreads();

    // ---- stage 1: hidden = silu(rfeat @ Wr1 + b) ----
    v2f a1[4];
    #pragma unroll
    for (int kk = 0; kk < 4; kk++) {
        const float* p = rfeat + (size_t)(e0 + lm) * N_RBF + kk * 4 + 2 * lg;
        a1[kk].x = p[0]; a1[kk].y = p[1];
    }
    #pragma unroll
    for (int nt = 0; nt < HR / 16; nt++) {
        int n0 = nt * 16;
        v8f acc = {0.f, 0.f, 0.f, 0.f, 0.f, 0.f, 0.f, 0.f};
        #pragma unroll
        for (int kk = 0; kk < 4; kk++) {
            const float* p = &wr1s[(kk * 4 + 2 * lg) * HR + n0 + lm];
            v2f b; b.x = p[0]; b.y = p[HR];
            acc = wmma4(a1[kk], b, acc);
        }
        float bias = br1s[n0 + lm];
        #pragma unroll
        for (int j = 0; j < 8; j++) {
            float h = acc[j] + bias;
            h = h / (1.0f + __expf(-h));            // silu
            hid[wave][(j + 8 * lg) * HR + n0 + lm] = h;
        }
    }
    asm volatile("s_wait_dscnt 0x0" ::: "memory");

    // ---- stage 2: w = hid @ Wr2  (16x64 @ 64x256), B from LDS ----
    v2f a2[16];
    #pragma unroll
    for (int kk = 0; kk < 16; kk++) {
        const float* p = &hid[wave][lm * HR + kk * 4 + 2 * lg];
        a2[kk].x = p[0]; a2[kk].y = p[1];
    }
    for (int nt = 0; nt < (2 * C) / 16; nt++) {
        int n0 = nt * 16;
        v8f acc = {0.f, 0.f, 0.f, 0.f, 0.f, 0.f, 0.f, 0.f};
        #pragma unroll
        for (int kk = 0; kk < 16; kk++) {
            const float* p = &wr2s[(kk * 4 + 2 * lg) * (2 * C) + n0 + lm];
            v2f b; b.x = p[0]; b.y = p[2 * C];
            acc = wmma4(a2[kk], b, acc);
        }
        #pragma unroll
        for (int j = 0; j < 8; j++)
            wt[wave][(j + 8 * lg) * (2 * C) + n0 + lm] = acc[j];
    }
    asm volatile("s_wait_dscnt 0x0" ::: "memory");

    // ---- stage 3: masked message scatter ----
    for (int e = 0; e < 16; e++) {
        int edge = e0 + e;
        int s = src[edge], dd = dst[edge];
        if (s == dd) continue;                       // wave-uniform branch
        float ux = u[(size_t)edge * 3 + 0];
        float uy = u[(size_t)edge * 3 + 1];
        float uz = u[(size_t)edge * 3 + 2];
        #pragma unroll
        for (int ci = 0; ci < 4; ci++) {
            int c = lane + ci * 32;
            float w_s = wt[wave][e * (2 * C) + c];
            float w_v = wt[wave][e * (2 * C) + C + c];
            float hsv = hs[(size_t)s * C + c];
            atomicAdd(&agg_s[(size_t)dd * C + c], w_s * hsv);
            size_t bi = ((size_t)s * C + c) * 3;
            size_t bo = ((size_t)dd * C + c) * 3;
            atomicAdd(&agg_v[bo + 0], w_v * (hv[bi + 0] + hsv * ux));
            atomicAdd(&agg_v[bo + 1], w_v * (hv[bi + 1] + hsv * uy));
            atomicAdd(&agg_v[bo + 2], w_v * (hv[bi + 2] + hsv * uz));
        }
    }
}

// ---------------------------------------------------------------------------
// 5) s_out = A@B0 + A^2@B1 + A^3@B2 ; h_s = (acc ? h_s : 0) + s_out
// ---------------------------------------------------------------------------
__global__ __launch_bounds__(32) void gemm_poly_kernel(
    const float* __restrict__ A, const float* __restrict__ B0,
    const float* __restrict__ B1, const float* __restrict__ B2,
    float* __restrict__ hs, int accumulate) {
    int lane = threadIdx.x & 31, lm = lane & 15, lg = lane >> 4;
    int row0 = blockIdx.x * (16 * MBLK), n0 = blockIdx.y * 16;
    v8f acc[MBLK];
    #pragma unroll
    for (int m = 0; m < MBLK; m++)
        acc[m] = (v8f){0.f, 0.f, 0.f, 0.f, 0.f, 0.f, 0.f, 0.f};
    for (int kk = 0; kk < C / 4; kk++) {
        int k0 = kk * 4;
        v2f b0 = loadB(B0, C, k0, n0, lm, lg);
        v2f b1 = loadB(B1, C, k0, n0, lm, lg);
        v2f b2 = loadB(B2, C, k0, n0, lm, lg);
        #pragma unroll
        for (int m = 0; m < MBLK; m++) {
            v2f a  = loadA(A, C, row0 + 16 * m, k0, lm, lg);
            v2f a2 = a * a;
            v2f a3 = a2 * a;
            acc[m] = wmma4(a,  b0, acc[m]);
            acc[m] = wmma4(a2, b1, acc[m]);
            acc[m] = wmma4(a3, b2, acc[m]);
        }
    }
    #pragma unroll
    for (int m = 0; m < MBLK; m++)
        #pragma unroll
        for (int j = 0; j < 8; j++) {
            size_t idx = (size_t)(row0 + 16 * m + j + 8 * lg) * C + n0 + lm;
            float v = acc[m][j];
            if (accumulate) v += hs[idx];
            hs[idx] = v;
        }
}

// ---------------------------------------------------------------------------
// 6) vtmp[n,d,x] = sum_c agg_v[n,c,x] * Wv[c,d]   (A strided by 3 over k)
// ---------------------------------------------------------------------------
__global__ __launch_bounds__(32) void gemm_v_kernel(
    const float* __restrict__ aggv, const float* __restrict__ Bv,
    float* __restrict__ vtmp) {
    int lane = threadIdx.x & 31, lm = lane & 15, lg = lane >> 4;
    int row0 = blockIdx.x * (16 * MBLK), n0 = blockIdx.y * 16, x = blockIdx.z;
    v8f acc[MBLK];
    #pragma unroll
    for (int m = 0; m < MBLK; m++)
        acc[m] = (v8f){0.f, 0.f, 0.f, 0.f, 0.f, 0.f, 0.f, 0.f};
    for (int kk = 0; kk < C / 4; kk++) {
        int k0 = kk * 4;
        v2f b = loadB(Bv, C, k0, n0, lm, lg);
        #pragma unroll
        for (int m = 0; m < MBLK; m++) {
            size_t base = ((size_t)(row0 + 16 * m + lm) * C + k0 + 2 * lg) * 3 + x;
            v2f a; a.x = aggv[base]; a.y = aggv[base + 3];
            acc[m] = wmma4(a, b, acc[m]);
        }
    }
    #pragma unroll
    for (int m = 0; m < MBLK; m++)
        #pragma unroll
        for (int j = 0; j < 8; j++)
            vtmp[((size_t)(row0 + 16 * m + j + 8 * lg) * C + n0 + lm) * 3 + x] =
                acc[m][j];
}

// ---------------------------------------------------------------------------
// 7) h_v update + write unnormalized feats into block_repr slab
// ---------------------------------------------------------------------------
__global__ __launch_bounds__(256) void v_ew_kernel(
    const float* __restrict__ aggs, const float* __restrict__ aggv,
    const float* __restrict__ vtmp, const float* __restrict__ wsvl,
    const float* __restrict__ hs, float* __restrict__ hv,
    float* __restrict__ blockrep, int l) {
    size_t idx = (size_t)blockIdx.x * 256 + threadIdx.x;  // n*C + c
    int n = (int)(idx >> 7);
    int c = (int)(idx & (C - 1));
    float as = aggs[idx];
    float w = wsvl[c];
    float* ob = blockrep + (size_t)n * (NL * 4 * C) + l * (4 * C);
    ob[c] = hs[idx];
    #pragma unroll
    for (int x = 0; x < 3; x++) {
        float v = vtmp[idx * 3 + x] + w * as * aggv[idx * 3 + x];
        if (l) v += hv[idx * 3 + x];
        hv[idx * 3 + x] = v;
        ob[C + c * 3 + x] = v;
    }
}

// ---------------------------------------------------------------------------
// 8) normalize block rows (1024) + atomic pool into graph accumulator
// ---------------------------------------------------------------------------
__global__ __launch_bounds__(256) void norm_block_kernel(
    float* __restrict__ blockrep, const int* __restrict__ batch_id,
    float* __restrict__ g) {
    __shared__ float red[256];
    int n = blockIdx.x;
    float* row = blockrep + (size_t)n * 1024;
    float ss = 0.f;
    for (int j = threadIdx.x; j < 1024; j += 256) { float v = row[j]; ss += v * v; }
    red[threadIdx.x] = ss; __syncthreads();
    for (int s = 128; s > 0; s >>= 1) {
        if (threadIdx.x < s) red[threadIdx.x] += red[threadIdx.x + s];
        __syncthreads();
    }
    float sc = 1.0f / fmaxf(sqrtf(red[0]), 1e-12f);
    int b = batch_id[n];
    for (int j = threadIdx.x; j < 1024; j += 256) {
        float v = row[j] * sc;
        row[j] = v;
        atomicAdd(&g[(size_t)b * 1024 + j], v);
    }
}

__global__ __launch_bounds__(256) void norm_graph_kernel(
    const float* __restrict__ g, float* __restrict__ graphrep) {
    __shared__ float red[256];
    int n = blockIdx.x;
    const float* row = g + (size_t)n * 1024;
    float ss = 0.f;
    for (int j = threadIdx.x; j < 1024; j += 256) { float v = row[j]; ss += v * v; }
    red[threadIdx.x] = ss; __syncthreads();
    for (int s = 128; s > 0; s >>= 1) {
        if (threadIdx.x < s) red[threadIdx.x] += red[threadIdx.x + s];
        __syncthreads();
    }
    float sc = 1.0f / fmaxf(sqrtf(red[0]), 1e-12f);
    for (int j = threadIdx.x; j < 1024; j += 256)
        graphrep[(size_t)n * 1024 + j] = row[j] * sc;
}

// ---------------------------------------------------------------------------
extern "C" void kernel_launch(void* const* d_in, const int* in_sizes, int n_in,
                              void* d_out, int out_size, void* d_ws,
                              size_t ws_size, hipStream_t stream) {
    (void)in_sizes; (void)n_in; (void)out_size; (void)ws_size;
    const float* H       = (const float*)d_in[0];
    const float* Z       = (const float*)d_in[1];
    const float* W_embed = (const float*)d_in[2];
    const float* Wr1     = (const float*)d_in[3];
    const float* br1     = (const float*)d_in[4];
    const float* Wr2     = (const float*)d_in[5];
    const float* Ws      = (const float*)d_in[6];
    const float* Wv      = (const float*)d_in[7];
    const float* wsv     = (const float*)d_in[8];
    const int* block_id  = (const int*)d_in[9];
    const int* batch_id  = (const int*)d_in[10];
    const int* edges     = (const int*)d_in[11];
    const int* src = edges;
    const int* dst = edges + NE;

    float* out      = (float*)d_out;
    float* Hb       = out;                                   // N_BLK*C
    float* blockrep = out + (size_t)N_BLK * C;               // N_BLK*1024
    float* graphrep = blockrep + (size_t)N_BLK * 1024;       // N_GR*1024

    // workspace layout (floats)
    float* ws    = (float*)d_ws;
    float* cnt   = ws;                        // 20000
    float* Zb    = cnt   + N_BLK;             // 60000   (contiguous w/ cnt)
    float* u     = Zb    + (size_t)N_BLK * 3; // E*3
    float* rfeat = u     + (size_t)NE * 3;    // E*16
    float* h_s   = rfeat + (size_t)NE * N_RBF;       // N_BLK*C
    float* h_v   = h_s   + (size_t)N_BLK * C;        // N_BLK*C*3
    float* agg_s = h_v   + (size_t)N_BLK * C * 3;    // N_BLK*C   (contig w/ agg_v)
    float* agg_v = agg_s + (size_t)N_BLK * C;        // N_BLK*C*3
    float* vtmp  = agg_v + (size_t)N_BLK * C * 3;    // N_BLK*C*3
    float* g     = vtmp  + (size_t)N_BLK * C * 3;    // N_GR*1024

    // per-call initialization (graph-capture safe)
    hipMemsetAsync(Hb, 0, (size_t)N_BLK * C * sizeof(float), stream);
    hipMemsetAsync(cnt, 0, (size_t)(N_BLK + N_BLK * 3) * sizeof(float), stream);
    hipMemsetAsync(h_v, 0, (size_t)N_BLK * C * 3 * sizeof(float), stream);
    hipMemsetAsync(g, 0, (size_t)N_GR * 1024 * sizeof(float), stream);

    seg_acc_kernel<<<N_ATOMS, 128, 0, stream>>>(H, Z, block_id, Hb, Zb, cnt);
    seg_norm_kernel<<<N_BLK, 128, 0, stream>>>(Hb, Zb, cnt);
    edge_kernel<<<(NE + 255) / 256, 256, 0, stream>>>(Zb, src, dst, u, rfeat);
    gemm_embed_kernel<<<dim3(N_BLK / (16 * MBLK), C / 16), 32, 0, stream>>>(
        Hb, W_embed, h_s);

    for (int l = 0; l < NL; l++) {
        hipMemsetAsync(agg_s, 0, (size_t)N_BLK * C * 4 * sizeof(float), stream);
        msg_kernel<<<NE / (16 * MSG_WAVES), 32 * MSG_WAVES, 0, stream>>>(
            rfeat, u, src, dst,
            Wr1 + (size_t)l * N_RBF * HR, br1 + (size_t)l * HR,
            Wr2 + (size_t)l * HR * 2 * C, h_s, h_v, agg_s, agg_v);
        gemm_poly_kernel<<<dim3(N_BLK / (16 * MBLK), C / 16), 32, 0, stream>>>(
            agg_s,
            Ws + ((size_t)l * 3 + 0) * C * C,
            Ws + ((size_t)l * 3 + 1) * C * C,
            Ws + ((size_t)l * 3 + 2) * C * C, h_s, l);
        gemm_v_kernel<<<dim3(N_BLK / (16 * MBLK), C / 16, 3), 32, 0, stream>>>(
            agg_v, Wv + (size_t)l * C * C, vtmp);
        v_ew_kernel<<<(N_BLK * C) / 256, 256, 0, stream>>>(
            agg_s, agg_v, vtmp, wsv + (size_t)l * C, h_s, h_v, blockrep, l);
    }

    norm_block_kernel<<<N_BLK, 256, 0, stream>>>(blockrep, batch_id, g);
    norm_graph_kernel<<<N_GR, 256, 0, stream>>>(g, graphrep);
}